// SequentialGeometricUpdate_40329742909863
// MI455X (gfx1250) — compile-verified
//
#include <hip/hip_runtime.h>
#include <math.h>

// ---------------- problem constants (match reference) ----------------
#define KNN   16
#define NPTS  4096
#define NFPS  1024
#define EPSN  1e-12f

typedef __attribute__((ext_vector_type(2))) float v2f;
typedef __attribute__((ext_vector_type(8))) float v8f;

// ---------------- small helpers ----------------
__device__ __forceinline__ float3 ld3(const float* p) {
    return make_float3(p[0], p[1], p[2]);
}
__device__ __forceinline__ void st3(float* p, float3 v) {
    p[0] = v.x; p[1] = v.y; p[2] = v.z;
}
__device__ __forceinline__ float dot3(float3 a, float3 b) {
    return a.x * b.x + a.y * b.y + a.z * b.z;
}
__device__ __forceinline__ float3 cross3(float3 a, float3 b) {
    return make_float3(a.y * b.z - a.z * b.y,
                       a.z * b.x - a.x * b.z,
                       a.x * b.y - a.y * b.x);
}

// F.normalize(dim=-1) semantics: v / max(|v|, eps)
__device__ __forceinline__ float3 normalize_eps(float3 v) {
    float n = sqrtf(dot3(v, v));
    float inv = 1.0f / fmaxf(n, EPSN);
    return make_float3(v.x * inv, v.y * inv, v.z * inv);
}

// Smallest-eigenvalue eigenvector of symmetric 3x3 [[a,d,e],[d,b,f],[e,f,c]]
// (trigonometric eigenvalue formula + row-cross-product eigenvector).
__device__ __forceinline__ float3 smallest_evec(float a, float b, float c,
                                                float d, float e, float f) {
    float p1 = d * d + e * e + f * f;
    if (p1 < 1e-30f) {   // already diagonal
        if (a <= b && a <= c) return make_float3(1.f, 0.f, 0.f);
        if (b <= c)           return make_float3(0.f, 1.f, 0.f);
        return make_float3(0.f, 0.f, 1.f);
    }
    float q  = (a + b + c) * (1.0f / 3.0f);
    float aa = a - q, bb = b - q, cc = c - q;
    float p2 = aa * aa + bb * bb + cc * cc + 2.0f * p1;
    float p  = sqrtf(p2 * (1.0f / 6.0f));
    float ip = 1.0f / p;
    float b00 = aa * ip, b11 = bb * ip, b22 = cc * ip;
    float b01 = d * ip,  b02 = e * ip,  b12 = f * ip;
    float detB = b00 * (b11 * b22 - b12 * b12)
               - b01 * (b01 * b22 - b12 * b02)
               + b02 * (b01 * b12 - b11 * b02);
    float r = fminf(1.0f, fmaxf(-1.0f, 0.5f * detB));
    float phi = acosf(r) * (1.0f / 3.0f);
    float lam = q + 2.0f * p * cosf(phi + 2.0943951023931953f);  // smallest eig

    float3 r0 = make_float3(a - lam, d, e);
    float3 r1 = make_float3(d, b - lam, f);
    float3 r2 = make_float3(e, f, c - lam);
    float3 v0 = cross3(r0, r1), v1 = cross3(r0, r2), v2 = cross3(r1, r2);
    float n0 = dot3(v0, v0), n1 = dot3(v1, v1), n2 = dot3(v2, v2);
    float3 v = v0; float nb = n0;
    if (n1 > nb) { v = v1; nb = n1; }
    if (n2 > nb) { v = v2; nb = n2; }
    float inv = rsqrtf(fmaxf(nb, 1e-30f));
    v = make_float3(v.x * inv, v.y * inv, v.z * inv);
    // deterministic sign: largest-|.| component made positive
    float ax = fabsf(v.x), ay = fabsf(v.y), az = fabsf(v.z);
    float m = (ax > ay) ? ((ax > az) ? v.x : v.z) : ((ay > az) ? v.y : v.z);
    if (m < 0.f) v = make_float3(-v.x, -v.y, -v.z);
    return v;
}

// ---------------- wave-level WMMA KNN core ----------------
// One wave32 handles 32 consecutive queries (= two 16x16 WMMA M-tiles).
// For each 16-point slab: D = Q(16x4,fp32) x P(4x16,fp32) via
// V_WMMA_F32_16X16X4_F32 (K dim = xyz + zero pad), then
// dist = |q|^2 + |p|^2 - 2*dot, per-lane branchless sorted top-16 insertion.
//
// fp32 A-matrix 16x4 layout (ISA 7.12.2): lanes 0-15 hold K=0,1 in v[0],v[1];
// lanes 16-31 hold K=2,3. B (4x16) analogous with N striped over lanes.
// C/D: lanes 0-15 -> M = r, N = lane ; lanes 16-31 -> M = r+8, N = lane-16.
__device__ __forceinline__ void wave_knn16(
    const float* __restrict__ qb,   // queries  [Mq,3] batch base
    const float* __restrict__ pb,   // points   [NPTS,3] batch base
    int qbase, int lane,
    float* __restrict__ dotT,       // LDS, 32*17 floats (this wave's slice)
    float* __restrict__ ppT,        // LDS, 16 floats    (this wave's slice)
    float3& q_own, float bestD[KNN], int bestI[KNN]) {
    const int sub  = lane & 15;
    const int half = lane >> 4;

    q_own = ld3(qb + (size_t)(qbase + lane) * 3);
    const float qq = dot3(q_own, q_own);

    float3 qa0 = ld3(qb + (size_t)(qbase + sub) * 3);
    float3 qa1 = ld3(qb + (size_t)(qbase + 16 + sub) * 3);
    v2f a0, a1;
    a0.x = half ? qa0.z : qa0.x;  a0.y = half ? 0.f : qa0.y;
    a1.x = half ? qa1.z : qa1.x;  a1.y = half ? 0.f : qa1.y;

#pragma unroll
    for (int k = 0; k < KNN; ++k) { bestD[k] = 3.4e38f; bestI[k] = 0; }

    for (int n0 = 0; n0 < NPTS; n0 += 16) {
        float3 pn = ld3(pb + (size_t)(n0 + sub) * 3);
        v2f bv;
        bv.x = half ? pn.z : pn.x;  bv.y = half ? 0.f : pn.y;
        // both half-waves store the identical |p|^2 (benign duplicate write):
        // avoids EXEC save/restore churn in the hot loop.
        ppT[sub] = dot3(pn, pn);

        v8f c0 = {}; v8f c1 = {};
        c0 = __builtin_amdgcn_wmma_f32_16x16x4_f32(false, a0, false, bv,
                                                   (short)0, c0, false, false);
        c1 = __builtin_amdgcn_wmma_f32_16x16x4_f32(false, a1, false, bv,
                                                   (short)0, c1, false, false);

        // transpose accumulator layout through LDS (row 17 stride: bank pad)
#pragma unroll
        for (int r = 0; r < 8; ++r) {
            dotT[(r + half * 8) * 17 + sub]      = c0[r];
            dotT[(16 + r + half * 8) * 17 + sub] = c1[r];
        }
        asm volatile("s_wait_dscnt 0" ::: "memory");  // DS writes visible to wave

        // lane owns query row `lane`; scan 16 candidates of this slab
#pragma unroll
        for (int j = 0; j < 16; ++j) {
            float dcand = qq + ppT[j] - 2.0f * dotT[lane * 17 + j];
            if (dcand < bestD[KNN - 1]) {     // cheap early reject (rare accept)
                float cd = dcand; int ci = n0 + j;
#pragma unroll
                for (int s = 0; s < KNN; ++s) {
                    // branchless compare-exchange: v_cmp + min/max/cndmask,
                    // no EXEC masking, no register-array copies.
                    const bool  lt = cd < bestD[s];
                    const float dl = lt ? cd : bestD[s];
                    const float dh = lt ? bestD[s] : cd;
                    const int   il = lt ? ci : bestI[s];
                    const int   ih = lt ? bestI[s] : ci;
                    bestD[s] = dl; bestI[s] = il;
                    cd = dh; ci = ih;
                }
            }
        }
        // in-wave DS ops are in-order: next slab's DS writes cannot pass these reads
    }
}

// ---------------- kernel 1: KNN(self) + covariance eigen-normal ----------------
__global__ __launch_bounds__(256) void k1_knn_normals(
    const float* __restrict__ xyz, int totalWaves,
    int* __restrict__ idx1, float* __restrict__ xmean, float* __restrict__ normal) {
    __shared__ float sdot[8][32 * 17];
    __shared__ float spp[8][16];
    const int wv = threadIdx.x >> 5, lane = threadIdx.x & 31;
    const int gw = blockIdx.x * 8 + wv;
    if (gw >= totalWaves) return;                 // whole-wave uniform exit
    const int wavesPerBatch = NPTS / 32;
    const int b = gw / wavesPerBatch;
    const int qbase = (gw % wavesPerBatch) * 32;
    const float* base = xyz + (size_t)b * NPTS * 3;

    float bestD[KNN]; int bestI[KNN]; float3 q;
    wave_knn16(base, base, qbase, lane, sdot[wv], spp[wv], q, bestD, bestI);

    // neighbor mean (pass 1) and covariance (pass 2) — gathers hit L0/L2
    float3 mu = make_float3(0.f, 0.f, 0.f);
#pragma unroll
    for (int k = 0; k < KNN; ++k) {
        float3 p = ld3(base + (size_t)bestI[k] * 3);
        mu.x += p.x; mu.y += p.y; mu.z += p.z;
    }
    const float inv16 = 1.0f / 16.0f;
    mu = make_float3(mu.x * inv16, mu.y * inv16, mu.z * inv16);

    float cxx = 0.f, cyy = 0.f, czz = 0.f, cxy = 0.f, cxz = 0.f, cyz = 0.f;
#pragma unroll
    for (int k = 0; k < KNN; ++k) {
        float3 p = ld3(base + (size_t)bestI[k] * 3);
        float x = p.x - mu.x, y = p.y - mu.y, z = p.z - mu.z;
        cxx += x * x; cyy += y * y; czz += z * z;
        cxy += x * y; cxz += x * z; cyz += y * z;
    }
    float3 nrm = smallest_evec(cxx, cyy, czz, cxy, cxz, cyz);

    const size_t o = (size_t)b * NPTS + (qbase + lane);
    st3(xmean  + o * 3, mu);
    st3(normal + o * 3, nrm);
#pragma unroll
    for (int k = 0; k < KNN; ++k) idx1[o * KNN + k] = bestI[k];
}

// ---------------- kernel 2: neighbor-normal mean + normal-projection update ----
__global__ __launch_bounds__(256) void k2_normal_update(
    const float* __restrict__ xyz, const int* __restrict__ idx1,
    const float* __restrict__ xmean, const float* __restrict__ normal,
    float* __restrict__ nmean, float* __restrict__ xyznu, int total) {
    int i = blockIdx.x * blockDim.x + threadIdx.x;
    if (i >= total) return;
    int b = i / NPTS;
    const float* nb = normal + (size_t)b * NPTS * 3;
    const int*   id = idx1 + (size_t)i * KNN;
    float3 s = make_float3(0.f, 0.f, 0.f);
#pragma unroll
    for (int k = 0; k < KNN; ++k) {
        float3 v = ld3(nb + (size_t)id[k] * 3);
        s.x += v.x; s.y += v.y; s.z += v.z;
    }
    const float inv16 = 1.0f / 16.0f;
    float3 nm = normalize_eps(make_float3(s.x * inv16, s.y * inv16, s.z * inv16));
    float3 x  = ld3(xyz + (size_t)i * 3);
    float3 mu = ld3(xmean + (size_t)i * 3);
    float3 dl = make_float3(x.x - mu.x, x.y - mu.y, x.z - mu.z);
    float dp = dot3(nm, dl);
    st3(nmean + (size_t)i * 3, nm);
    st3(xyznu + (size_t)i * 3,
        make_float3(x.x - dp * nm.x, x.y - dp * nm.y, x.z - dp * nm.z));
}

// ---------------- kernel 3: farthest point sampling (1 block per batch) -------
__global__ __launch_bounds__(1024) void k3_fps(
    const float* __restrict__ xyznu, float* __restrict__ xyzfps) {
    const int b = blockIdx.x;
    const float* pts = xyznu + (size_t)b * NPTS * 3;
    float* outp = xyzfps + (size_t)b * NFPS * 3;
    const int tid = threadIdx.x, lane = tid & 31, wv = tid >> 5;
    __shared__ float sval[32];
    __shared__ int   sidx[32];
    __shared__ int   ssel;
    const int PT = NPTS / 1024;        // 4 points per thread, strided ownership
    float  mind[PT];
    float3 own[PT];
#pragma unroll
    for (int j = 0; j < PT; ++j) {
        mind[j] = 3.4e38f;
        own[j]  = ld3(pts + (size_t)(tid + j * 1024) * 3);
    }
    int sel = 0;
    for (int step = 0; step < NFPS; ++step) {
        float3 pc = ld3(pts + (size_t)sel * 3);   // broadcast via cache
        if (tid == 0) st3(outp + (size_t)step * 3, pc);
        float bv = -1.f; int bi = 0;
#pragma unroll
        for (int j = 0; j < PT; ++j) {
            float dx = own[j].x - pc.x, dy = own[j].y - pc.y, dz = own[j].z - pc.z;
            float dd = dx * dx + dy * dy + dz * dz;
            mind[j] = fminf(mind[j], dd);
            if (mind[j] > bv) { bv = mind[j]; bi = tid + j * 1024; }
        }
        // wave32 butterfly argmax (lowest index wins ties, like jnp.argmax)
#pragma unroll
        for (int off = 16; off >= 1; off >>= 1) {
            float ov = __shfl_xor(bv, off, 32);
            int   oi = __shfl_xor(bi, off, 32);
            if (ov > bv || (ov == bv && oi < bi)) { bv = ov; bi = oi; }
        }
        if (lane == 0) { sval[wv] = bv; sidx[wv] = bi; }
        __syncthreads();
        if (wv == 0) {
            float v = sval[lane]; int ii = sidx[lane];
#pragma unroll
            for (int off = 16; off >= 1; off >>= 1) {
                float ov = __shfl_xor(v, off, 32);
                int   oi = __shfl_xor(ii, off, 32);
                if (ov > v || (ov == v && oi < ii)) { v = ov; ii = oi; }
            }
            if (lane == 0) ssel = ii;
        }
        __syncthreads();
        sel = ssel;
    }
}

// ---------------- kernel 4: KNN(fps vs nu) + tangential smoothing -> output ---
__global__ __launch_bounds__(256) void k4_knn_out(
    const float* __restrict__ xyzfps, const float* __restrict__ xyznu,
    const float* __restrict__ nmean, int totalWaves, float* __restrict__ outp) {
    __shared__ float sdot[8][32 * 17];
    __shared__ float spp[8][16];
    const int wv = threadIdx.x >> 5, lane = threadIdx.x & 31;
    const int gw = blockIdx.x * 8 + wv;
    if (gw >= totalWaves) return;
    const int wavesPerBatch = NFPS / 32;
    const int b = gw / wavesPerBatch;
    const int qbase = (gw % wavesPerBatch) * 32;
    const float* qb  = xyzfps + (size_t)b * NFPS * 3;
    const float* pb  = xyznu  + (size_t)b * NPTS * 3;
    const float* nmb = nmean  + (size_t)b * NPTS * 3;

    float bestD[KNN]; int bestI[KNN]; float3 q;
    wave_knn16(qb, pb, qbase, lane, sdot[wv], spp[wv], q, bestD, bestI);

    float3 mu = make_float3(0.f, 0.f, 0.f);
    float3 ns = make_float3(0.f, 0.f, 0.f);
#pragma unroll
    for (int k = 0; k < KNN; ++k) {
        float3 p = ld3(pb + (size_t)bestI[k] * 3);
        float3 n = ld3(nmb + (size_t)bestI[k] * 3);
        mu.x += p.x; mu.y += p.y; mu.z += p.z;
        ns.x += n.x; ns.y += n.y; ns.z += n.z;
    }
    const float inv16 = 1.0f / 16.0f;
    mu = make_float3(mu.x * inv16, mu.y * inv16, mu.z * inv16);
    float3 n2 = normalize_eps(make_float3(ns.x * inv16, ns.y * inv16, ns.z * inv16));
    float3 dl = make_float3(q.x - mu.x, q.y - mu.y, q.z - mu.z);
    float dp = dot3(n2, dl);
    // out = q - (delta - (n.delta)n) = x_mean2 + (n.delta) n
    const size_t o = ((size_t)b * NFPS + qbase + lane) * 3;
    outp[o + 0] = mu.x + dp * n2.x;
    outp[o + 1] = mu.y + dp * n2.y;
    outp[o + 2] = mu.z + dp * n2.z;
}

// ---------------- host launcher ----------------
extern "C" void kernel_launch(void* const* d_in, const int* in_sizes, int n_in,
                              void* d_out, int out_size, void* d_ws, size_t ws_size,
                              hipStream_t stream) {
    const float* xyz = (const float*)d_in[0];
    const int B = in_sizes[0] / (NPTS * 3);   // 4
    float* outp = (float*)d_out;

    // workspace carving (all fp32/int32), ~1.85 MB total for B=4
    char* ws = (char*)d_ws;
    size_t off = 0;
    int*   idx1   = (int*)(ws + off);  off += (size_t)B * NPTS * KNN * sizeof(int);
    float* xmean  = (float*)(ws + off); off += (size_t)B * NPTS * 3 * sizeof(float);
    float* normal = (float*)(ws + off); off += (size_t)B * NPTS * 3 * sizeof(float);
    float* nmean  = (float*)(ws + off); off += (size_t)B * NPTS * 3 * sizeof(float);
    float* xyznu  = (float*)(ws + off); off += (size_t)B * NPTS * 3 * sizeof(float);
    float* xyzfps = (float*)(ws + off); off += (size_t)B * NFPS * 3 * sizeof(float);
    (void)ws_size; (void)n_in; (void)out_size;

    const int waves1 = B * NPTS / 32;                    // 512
    k1_knn_normals<<<(waves1 + 7) / 8, 256, 0, stream>>>(xyz, waves1, idx1, xmean, normal);

    const int tot = B * NPTS;
    k2_normal_update<<<(tot + 255) / 256, 256, 0, stream>>>(xyz, idx1, xmean, normal,
                                                            nmean, xyznu, tot);

    k3_fps<<<B, 1024, 0, stream>>>(xyznu, xyzfps);

    const int waves4 = B * NFPS / 32;                    // 128
    k4_knn_out<<<(waves4 + 7) / 8, 256, 0, stream>>>(xyzfps, xyznu, nmean, waves4, outp);
}